// MessageAggregator_74440373174623
// MI455X (gfx1250) — compile-verified
//
#include <hip/hip_runtime.h>

typedef __attribute__((ext_vector_type(2))) float v2f;
typedef __attribute__((ext_vector_type(8))) float v8f;

#define HH 4
#define DD 64
#define LSTR 68   // padded LDS row stride (floats): 272B rows -> float4-aligned, bank = 4*row+col

__device__ __forceinline__ void atomicMaxF(float* addr, float v) {
    // monotone bit-trick float max (works across mixed signs; init = -inf bits)
    if (v >= 0.0f) atomicMax((int*)addr, __float_as_int(v));
    else           atomicMin((unsigned int*)addr, __float_as_uint(v));
}

// Cooperative init of a 16x64 weight tile (rows >= HH zero-filled), padded stride.
// Each LDS word written by exactly one thread -> race-free before the barrier.
__device__ __forceinline__ void stage_weights(float* wlds, const float* __restrict__ W, int tid) {
#pragma unroll
    for (int j = 0; j < 5; ++j) {
        int x = tid + j * 256;
        if (x < 16 * LSTR) {
            int row = x / LSTR;
            int col = x - row * LSTR;
            wlds[x] = (row < HH && col < DD) ? W[row * DD + col] : 0.0f;
        }
    }
}

// ---------------- Phase 1: s1[R,4] = X[R,64] @ W[4,64]^T  (WMMA f32 16x16x4) --------------
__global__ __launch_bounds__(256) void k_rowscore_wmma(const float* __restrict__ X,
                                                       const float* __restrict__ W,
                                                       float* __restrict__ out, int R) {
    __shared__ float tileLds[8][16 * LSTR];
    __shared__ float wlds[16 * LSTR];
    const int tid = threadIdx.x;
    const int wave = tid >> 5;
    const int lane = tid & 31;

    stage_weights(wlds, W, tid);
    __syncthreads();

    const long rowbase = ((long)blockIdx.x * 8 + wave) * 16;
    if (rowbase + 16 > (long)R) return;   // uniform per wave, after the only barrier

    // coalesced copy of a contiguous 16x64 f32 tile into padded LDS
    float* tile = tileLds[wave];
    const float* src = X + rowbase * DD;
#pragma unroll
    for (int j = 0; j < 8; ++j) {
        int f = j * 128 + lane * 4;
        int r = f >> 6, c = f & 63;
        float4 v = *(const float4*)(src + f);
        *(float4*)(tile + r * LSTR + c) = v;
    }

    const int m = lane & 15;       // A row / B col
    const int hf = lane >> 4;      // half selects K pair
    const int n = m;
    v8f acc = {};
#pragma unroll
    for (int k0 = 0; k0 < DD; k0 += 4) {
        const int k = k0 + 2 * hf;
        v2f a, b;
        a.x = tile[m * LSTR + k];
        a.y = tile[m * LSTR + k + 1];
        b.x = wlds[n * LSTR + k];        // rows >= HH are zero -> unconditional b64 load
        b.y = wlds[n * LSTR + k + 1];
        acc = __builtin_amdgcn_wmma_f32_16x16x4_f32(false, a, false, b, (short)0, acc, false, false);
    }
    if (n < HH) {
#pragma unroll
        for (int v = 0; v < 8; ++v) {
            const int M = v + 8 * hf;
            out[(rowbase + M) * HH + n] = acc[v];
        }
    }
}

// ---- Phase 2: a[e,h] = leaky_relu(s1[idx[e],h] + emb[e,:]·W2[h,:]); atomic segmax ----
__global__ __launch_bounds__(256) void k_edgescore_wmma(const float* __restrict__ EMB,
                                                        const float* __restrict__ W2,
                                                        const float* __restrict__ s1,
                                                        const int* __restrict__ idx,
                                                        float* __restrict__ a_out,
                                                        float* __restrict__ segmax, int E) {
    __shared__ float tileLds[8][16 * LSTR];
    __shared__ float wlds[16 * LSTR];
    const int tid = threadIdx.x;
    const int wave = tid >> 5;
    const int lane = tid & 31;

    stage_weights(wlds, W2, tid);
    __syncthreads();

    const long ebase = ((long)blockIdx.x * 8 + wave) * 16;
    if (ebase + 16 > (long)E) return;

    float* tile = tileLds[wave];
    const float* src = EMB + ebase * DD;
#pragma unroll
    for (int j = 0; j < 8; ++j) {
        int f = j * 128 + lane * 4;
        int r = f >> 6, c = f & 63;
        float4 v = *(const float4*)(src + f);
        *(float4*)(tile + r * LSTR + c) = v;
    }

    const int m = lane & 15;
    const int hf = lane >> 4;
    const int n = m;
    v8f acc = {};
#pragma unroll
    for (int k0 = 0; k0 < DD; k0 += 4) {
        const int k = k0 + 2 * hf;
        v2f a, b;
        a.x = tile[m * LSTR + k];
        a.y = tile[m * LSTR + k + 1];
        b.x = wlds[n * LSTR + k];
        b.y = wlds[n * LSTR + k + 1];
        acc = __builtin_amdgcn_wmma_f32_16x16x4_f32(false, a, false, b, (short)0, acc, false, false);
    }
    if (n < HH) {
#pragma unroll
        for (int v = 0; v < 8; ++v) {
            const long e = ebase + v + 8 * hf;
            const int node = idx[e];
            float s = acc[v] + s1[(long)node * HH + n];
            s = (s > 0.0f) ? s : 0.2f * s;            // leaky_relu, alpha=0.2
            a_out[e * HH + n] = s;
            atomicMaxF(&segmax[(long)node * HH + n], s);
        }
    }
}

// ---------------- Phase 3: ex = exp(a - segmax[idx]); denom += ex --------------------------
__global__ __launch_bounds__(256) void k_exp(float* __restrict__ a, const int* __restrict__ idx,
                                             const float* __restrict__ segmax,
                                             float* __restrict__ denom, long EH) {
    long t = (long)blockIdx.x * 256 + threadIdx.x;
    if (t >= EH) return;
    long e = t >> 2; int h = t & 3;
    int node = idx[e];
    float ex = __expf(a[t] - segmax[(long)node * HH + h]);
    a[t] = ex;
    atomicAdd(&denom[(long)node * HH + h], ex);
}

// ---------------- Phase 3b: attn = ex / denom[idx] -----------------------------------------
__global__ __launch_bounds__(256) void k_norm(float* __restrict__ a, const int* __restrict__ idx,
                                              const float* __restrict__ denom, long EH) {
    long t = (long)blockIdx.x * 256 + threadIdx.x;
    if (t >= EH) return;
    long e = t >> 2; int h = t & 3;
    a[t] = a[t] / denom[(long)idx[e] * HH + h];
}

// ---------------- Phase 4: out[idx,h,d] += attn[e,h] * emb[e,d] ----------------------------
__global__ __launch_bounds__(256) void k_scatter(const float* __restrict__ emb,
                                                 const float* __restrict__ attn,
                                                 const int* __restrict__ idx,
                                                 float* __restrict__ out, long ED) {
    long t = (long)blockIdx.x * 256 + threadIdx.x;
    if (t >= ED) return;
    long e = t >> 6; int d = t & 63;
    float ev = emb[t];
    int node = idx[e];
    float* o = out + (long)node * (HH * DD) + d;
#pragma unroll
    for (int h = 0; h < HH; ++h)
        atomicAdd(o + h * DD, attn[e * HH + h] * ev);
}

// ---------------- Phase 5: ELU in place ----------------------------------------------------
__global__ __launch_bounds__(256) void k_elu(float* __restrict__ out, long n) {
    long t = (long)blockIdx.x * 256 + threadIdx.x;
    if (t >= n) return;
    float x = out[t];
    out[t] = (x > 0.0f) ? x : (__expf(x) - 1.0f);
}

// ---------------- Phase 0: init ------------------------------------------------------------
__global__ __launch_bounds__(256) void k_init(float* __restrict__ out, float* __restrict__ segmax,
                                              float* __restrict__ denom, long nOut, long nSeg) {
    long t = (long)blockIdx.x * 256 + threadIdx.x;
    if (t < nOut) out[t] = 0.0f;
    if (t < nSeg) { segmax[t] = -__builtin_inff(); denom[t] = 0.0f; }
}

extern "C" void kernel_launch(void* const* d_in, const int* in_sizes, int n_in,
                              void* d_out, int out_size, void* d_ws, size_t ws_size,
                              hipStream_t stream) {
    const float* emb  = (const float*)d_in[0];   // [E,64]
    const float* feat = (const float*)d_in[1];   // [N,64]
    const float* w1   = (const float*)d_in[2];   // [4,64]
    const float* w2   = (const float*)d_in[3];   // [4,64]
    const int*   idx  = (const int*)d_in[4];     // [E]
    const long E = in_sizes[4];
    const long N = in_sizes[1] / DD;
    float* out = (float*)d_out;

    float* s1     = (float*)d_ws;          // N*4
    float* a      = s1 + N * HH;           // E*4 (scores -> ex -> attn, in place)
    float* segmax = a + E * HH;            // N*4
    float* denom  = segmax + N * HH;       // N*4

    const long nOut = N * (long)(HH * DD);
    k_init<<<(unsigned)((nOut + 255) / 256), 256, 0, stream>>>(out, segmax, denom, nOut, N * HH);

    const long waves1 = (N + 15) / 16;
    k_rowscore_wmma<<<(unsigned)((waves1 + 7) / 8), 256, 0, stream>>>(feat, w1, s1, (int)N);

    const long waves2 = (E + 15) / 16;
    k_edgescore_wmma<<<(unsigned)((waves2 + 7) / 8), 256, 0, stream>>>(emb, w2, s1, idx, a, segmax, (int)E);

    const long EH = E * HH;
    k_exp<<<(unsigned)((EH + 255) / 256), 256, 0, stream>>>(a, idx, segmax, denom, EH);
    k_norm<<<(unsigned)((EH + 255) / 256), 256, 0, stream>>>(a, idx, denom, EH);

    const long ED = E * DD;
    k_scatter<<<(unsigned)((ED + 255) / 256), 256, 0, stream>>>(emb, a, idx, out, ED);
    k_elu<<<(unsigned)((nOut + 255) / 256), 256, 0, stream>>>(out, nOut);
}